// Quantizer_14353780703995
// MI455X (gfx1250) — compile-verified
//
#include <hip/hip_runtime.h>

// Elementwise stochastic fixed-point quantize (WL=8, FL=4):
//   q = clamp( floor(fma(x, 16, r)) * 0.0625, -8.0, 7.9375 )
// Memory-bound: 1.61 GB single-pass stream -> ~69 us floor at 23.3 TB/s.
// Hot kernel: guard-free, 32-bit offsets (SADDR+voffset form), B128 NT.

typedef float v4f __attribute__((ext_vector_type(4)));

#define TPB 256   // 8 wave32s per block
#define VPT 2     // float4 tiles per thread -> 2048 floats per block

__device__ __forceinline__ float quant1(float v, float rr) {
    // x/sigma (sigma = 2^-4) is an exact power-of-two scale, so fma(v,16,r)
    // is bit-identical to the reference's (v/sigma + r) single-rounded result.
    float q = floorf(fmaf(v, 16.0f, rr)) * 0.0625f;
    q = fmaxf(q, -8.0f);
    q = fminf(q, 7.9375f);
    return q;
}

// Fast path: launched when n % (TPB*VPT*4) == 0. No guards, no tail,
// 32-bit float4 indices (buffers < 4 GiB) -> saddr + 32-bit voffset addressing.
__global__ __launch_bounds__(TPB) void quant_fx_full(
    const v4f* __restrict__ x4,
    const v4f* __restrict__ r4,
    v4f* __restrict__ o4)
{
    const unsigned i0 = blockIdx.x * (unsigned)(TPB * VPT) + threadIdx.x;

    v4f vx[VPT], vr[VPT];
#pragma unroll
    for (int u = 0; u < VPT; ++u) {
        const unsigned i = i0 + (unsigned)(u * TPB);
        vx[u] = __builtin_nontemporal_load(&x4[i]);
        vr[u] = __builtin_nontemporal_load(&r4[i]);
    }

#pragma unroll
    for (int u = 0; u < VPT; ++u) {
        v4f q;
#pragma unroll
        for (int k = 0; k < 4; ++k) q[k] = quant1(vx[u][k], vr[u][k]);
        __builtin_nontemporal_store(q, &o4[i0 + (unsigned)(u * TPB)]);
    }
}

// Generic fallback for arbitrary n (guarded tiles + scalar tail).
__global__ __launch_bounds__(TPB) void quant_fx_generic(
    const v4f* __restrict__ x4,
    const v4f* __restrict__ r4,
    v4f* __restrict__ o4,
    const float* __restrict__ x,
    const float* __restrict__ r,
    float* __restrict__ out,
    long long n4,
    long long n)
{
    const long long base =
        (long long)blockIdx.x * (long long)(TPB * VPT) + (long long)threadIdx.x;

    v4f vx[VPT], vr[VPT];
    bool ok[VPT];
#pragma unroll
    for (int u = 0; u < VPT; ++u) {
        const long long i = base + (long long)u * TPB;
        ok[u] = (i < n4);
        if (ok[u]) {
            vx[u] = __builtin_nontemporal_load(&x4[i]);
            vr[u] = __builtin_nontemporal_load(&r4[i]);
        }
    }
#pragma unroll
    for (int u = 0; u < VPT; ++u) {
        if (ok[u]) {
            v4f q;
#pragma unroll
            for (int k = 0; k < 4; ++k) q[k] = quant1(vx[u][k], vr[u][k]);
            __builtin_nontemporal_store(q, &o4[base + (long long)u * TPB]);
        }
    }
    const long long tail = n4 * 4;
    if (blockIdx.x == 0) {
        for (long long j = tail + (long long)threadIdx.x; j < n; j += TPB) {
            out[j] = quant1(x[j], r[j]);
        }
    }
}

extern "C" void kernel_launch(void* const* d_in, const int* in_sizes, int n_in,
                              void* d_out, int out_size, void* d_ws, size_t ws_size,
                              hipStream_t stream) {
    (void)in_sizes; (void)n_in; (void)d_ws; (void)ws_size;

    const float* x = (const float*)d_in[0];
    const float* r = (const float*)d_in[1];
    float* out = (float*)d_out;

    const long long n = (long long)out_size;
    const long long n4 = n >> 2;
    const long long floatsPerBlock = (long long)TPB * VPT * 4;

    if (n > 0 && (n % floatsPerBlock) == 0 && n <= (1LL << 31)) {
        // Exact tiling (true for 32*1024*4096): guard-free kernel.
        const unsigned blocks = (unsigned)(n / floatsPerBlock);
        quant_fx_full<<<blocks, TPB, 0, stream>>>(
            (const v4f*)x, (const v4f*)r, (v4f*)out);
    } else {
        const long long tilesPerBlock = (long long)TPB * VPT;
        long long blocks = (n4 + tilesPerBlock - 1) / tilesPerBlock;
        if (blocks < 1) blocks = 1;
        quant_fx_generic<<<(dim3)(unsigned)blocks, TPB, 0, stream>>>(
            (const v4f*)x, (const v4f*)r, (v4f*)out, x, r, out, n4, n);
    }
}